// SpatialEncoding_38517266710631
// MI455X (gfx1250) — compile-verified
//
#include <hip/hip_runtime.h>
#include <hip/hip_bf16.h>
#include <stdint.h>

// Problem constants (fixed by the reference).
#define BATCH     128
#define BLOCK_SZ  72
#define MAX_PATH  5
#define NUM_NODES (BATCH * BLOCK_SZ)              // 9216
#define N_ELEMS   ((unsigned long long)NUM_NODES * (unsigned long long)NUM_NODES) // 84,934,656
#define N_CHUNK16 (N_ELEMS / 4ull)                // 21,233,664 16B chunks

#define FILL_BLOCKS  1296
#define FILL_THREADS 256
// 1296*256 = 331,776 threads; 21,233,664 / 331,776 = 64 chunks per thread exactly.

// ---------------------------------------------------------------------------
// Kernel 1: stream zeros over the whole 340MB output using the CDNA5 async
// LDS->global store path. Each lane owns a 16B zero chunk in LDS and issues
// global_store_async_from_lds_b128 (512B per wave per instruction, ASYNCcnt
// pipelined up to 63 deep, th:TH_STORE_NT so we don't thrash the 192MB L2
// with a 340MB stream). Drained with s_wait_asynccnt 0 before endpgm.
// ---------------------------------------------------------------------------
__global__ __launch_bounds__(FILL_THREADS)
void SpatialEncoding_fill_zero_async(float* __restrict__ out) {
    __shared__ __align__(16) float zbuf[FILL_THREADS * 4]; // 4KB of zeros
    zbuf[threadIdx.x * 4 + 0] = 0.0f;
    zbuf[threadIdx.x * 4 + 1] = 0.0f;
    zbuf[threadIdx.x * 4 + 2] = 0.0f;
    zbuf[threadIdx.x * 4 + 3] = 0.0f;
    __syncthreads();

    // Low 32 bits of a generic shared-aperture address == LDS byte offset.
    unsigned lds_addr = (unsigned)(uintptr_t)(&zbuf[threadIdx.x * 4]);

    const unsigned long long tid     = (unsigned long long)blockIdx.x * FILL_THREADS + threadIdx.x;
    const unsigned long long nthread = (unsigned long long)FILL_BLOCKS * FILL_THREADS;

    for (unsigned long long c = tid; c < N_CHUNK16; c += nthread) {
        unsigned byte_off = (unsigned)(c * 16ull); // < 2^29, fits signed 32-bit VGPR offset
        asm volatile(
            "global_store_async_from_lds_b128 %0, %1, %2 th:TH_STORE_NT"
            :
            : "v"(byte_off), "v"(lds_addr), "s"(out)
            : "memory");
    }
    asm volatile("s_wait_asynccnt 0" ::: "memory");
}

// ---------------------------------------------------------------------------
// Kernel 2: diagonal blocks. One thread per (batch, i, j) pair:
//   len = count(paths[bat,i,j,:] != -1)  (0..5), value = b[len],
//   out[(bat*72+i)*9216 + bat*72+j] = value.
// Consecutive threads -> consecutive j -> coalesced 288B row segments.
// ---------------------------------------------------------------------------
__global__ __launch_bounds__(256)
void SpatialEncoding_blocks(const int* __restrict__ paths,
                            const float* __restrict__ b,
                            float* __restrict__ out) {
    const unsigned idx = blockIdx.x * 256u + threadIdx.x; // 0 .. 663551
    const unsigned j   = idx % BLOCK_SZ;
    const unsigned t   = idx / BLOCK_SZ;
    const unsigned i   = t % BLOCK_SZ;
    const unsigned bat = t / BLOCK_SZ;

    const int* __restrict__ p = paths + (unsigned long long)idx * MAX_PATH;
    int len = 0;
#pragma unroll
    for (int k = 0; k < MAX_PATH; ++k)
        len += (p[k] != -1) ? 1 : 0;

    const float v = b[len];

    const unsigned long long base = (unsigned long long)bat * BLOCK_SZ;
    out[(base + i) * (unsigned long long)NUM_NODES + base + j] = v;
}

// ---------------------------------------------------------------------------
// Launch: fill (async NT stream of zeros) then scatter the diagonal blocks.
// Same stream -> kernels serialize, so the block values land after the fill.
// ---------------------------------------------------------------------------
extern "C" void kernel_launch(void* const* d_in, const int* in_sizes, int n_in,
                              void* d_out, int out_size, void* d_ws, size_t ws_size,
                              hipStream_t stream) {
    (void)in_sizes; (void)n_in; (void)out_size; (void)d_ws; (void)ws_size;

    // d_in[0] = x (float32, values unused by the reference output)
    const int*   paths = (const int*)d_in[1];   // [128,72,72,5] int
    const float* b     = (const float*)d_in[2]; // [6] float32
    float*       out   = (float*)d_out;         // [9216,9216] float32

    SpatialEncoding_fill_zero_async<<<FILL_BLOCKS, FILL_THREADS, 0, stream>>>(out);

    const unsigned npairs = BATCH * BLOCK_SZ * BLOCK_SZ; // 663,552
    SpatialEncoding_blocks<<<npairs / 256, 256, 0, stream>>>(paths, b, out);
}